// Jointer_75917841924365
// MI455X (gfx1250) — compile-verified
//
#include <hip/hip_runtime.h>

// RNN-T Jointer: out[b,t,u,v] = sum_d (enc[b,t,d]+dec[b,u,d])*W[v,d] + bias[v]
// Factored:     out = encW[b,t,v] + decW[b,u,v],  encW = enc@W^T + bias, decW = dec@W^T
// Phase 1: two small fp32 WMMA GEMMs (L2-resident operands, ~2.4 GFLOP total)
// Phase 2: bandwidth-bound broadcast add, 512 MB nontemporal b128 stores (~22 us @ 23.3 TB/s)

#define BD 4
#define TD 256
#define UD 128
#define DD 768
#define VD 1024

typedef float v2f __attribute__((ext_vector_type(2)));
typedef float v4f __attribute__((ext_vector_type(4)));
typedef float v8f __attribute__((ext_vector_type(8)));

// C[m, n] = sum_k X[m,k] * W[n,k] (+ bias[n])   X:[M,DD] row-major, W:[VD,DD] row-major
// One wave computes a 16x64 output tile with V_WMMA_F32_16X16X4_F32, K stepped by 4.
// A layout (16x4 f32): lane%16 = M, (lane/16)*2 + vgpr = K  -> contiguous float2 per lane.
// B layout (4x16 f32): lane%16 = N, (lane/16)*2 + vgpr = K  -> contiguous float2 of W row.
__global__ __launch_bounds__(256) void jointer_gemm_xwT(
    const float* __restrict__ X, const float* __restrict__ W,
    const float* __restrict__ bias, float* __restrict__ out, int M)
{
    const int lane      = threadIdx.x & 31;
    const int waveId    = (blockIdx.x * 256 + threadIdx.x) >> 5;
    const int nColTiles = VD / 64;                 // 16
    const int row0      = (waveId / nColTiles) * 16;
    const int col0      = (waveId % nColTiles) * 64;
    if (row0 >= M) return;                          // wave-uniform; EXEC stays all-1s

    const int mn   = lane & 15;
    const int koff = (lane >> 4) * 2;

    const float* aRow  = X + (size_t)(row0 + mn) * DD + koff;
    const float* bRow0 = W + (size_t)(col0 +  0 + mn) * DD + koff;
    const float* bRow1 = W + (size_t)(col0 + 16 + mn) * DD + koff;
    const float* bRow2 = W + (size_t)(col0 + 32 + mn) * DD + koff;
    const float* bRow3 = W + (size_t)(col0 + 48 + mn) * DD + koff;

    v8f acc0 = {}, acc1 = {}, acc2 = {}, acc3 = {};
    #pragma unroll 4
    for (int k = 0; k < DD; k += 4) {
        v2f a  = *(const v2f*)(aRow  + k);
        v2f b0 = *(const v2f*)(bRow0 + k);
        v2f b1 = *(const v2f*)(bRow1 + k);
        v2f b2 = *(const v2f*)(bRow2 + k);
        v2f b3 = *(const v2f*)(bRow3 + k);
        acc0 = __builtin_amdgcn_wmma_f32_16x16x4_f32(false, a, false, b0, (short)0, acc0, false, false);
        acc1 = __builtin_amdgcn_wmma_f32_16x16x4_f32(false, a, false, b1, (short)0, acc1, false, false);
        acc2 = __builtin_amdgcn_wmma_f32_16x16x4_f32(false, a, false, b2, (short)0, acc2, false, false);
        acc3 = __builtin_amdgcn_wmma_f32_16x16x4_f32(false, a, false, b3, (short)0, acc3, false, false);
    }

    // C/D layout (16x16 f32): vgpr r, lane<16 -> (M=r, N=lane); lane>=16 -> (M=r+8, N=lane-16)
    float bb0 = 0.f, bb1 = 0.f, bb2 = 0.f, bb3 = 0.f;
    if (bias) {
        bb0 = bias[col0 +  0 + mn];
        bb1 = bias[col0 + 16 + mn];
        bb2 = bias[col0 + 32 + mn];
        bb3 = bias[col0 + 48 + mn];
    }
    const int mBase = row0 + (lane >> 4) * 8;
    #pragma unroll
    for (int r = 0; r < 8; ++r) {
        float* o = out + (size_t)(mBase + r) * VD + col0 + mn;
        o[ 0] = acc0[r] + bb0;
        o[16] = acc1[r] + bb1;
        o[32] = acc2[r] + bb2;
        o[48] = acc3[r] + bb3;
    }
}

// out[((b*T+t)*U+u)*V+v] = encW[(b*T+t)*V+v] + decW[(b*U+u)*V+v]
// Block: one b, TC t's (encW held in regs), UC u's, full V via one float4/thread.
// Output is a pure 512 MB write stream -> nontemporal (TH=NT) b128 stores.
__global__ __launch_bounds__(256) void jointer_bcast_add(
    const float* __restrict__ encW, const float* __restrict__ decW,
    float* __restrict__ out)
{
    const int TC = 4, UC = 32;
    int blk = blockIdx.x;
    const int uBlocks = UD / UC;                    // 4
    const int tBlocks = TD / TC;                    // 64
    const int ub = blk % uBlocks; blk /= uBlocks;
    const int tb = blk % tBlocks; blk /= tBlocks;
    const int b  = blk;                             // 0..B-1
    const int v  = (int)threadIdx.x * 4;            // 256 threads cover V=1024

    const int t0 = tb * TC;
    v4f e[TC];
    #pragma unroll
    for (int i = 0; i < TC; ++i)
        e[i] = *(const v4f*)(encW + (size_t)(b * TD + t0 + i) * VD + v);

    const float* dptr   = decW + (size_t)(b * UD + ub * UC) * VD + v;
    float*       obase  = out  + ((size_t)(b * TD + t0) * UD + ub * UC) * VD + v;

    for (int u = 0; u < UC; ++u) {
        v4f d = *(const v4f*)(dptr + (size_t)u * VD);
        #pragma unroll
        for (int i = 0; i < TC; ++i) {
            v4f o = e[i] + d;
            __builtin_nontemporal_store(
                o, (v4f*)(obase + (size_t)i * (UD * VD) + (size_t)u * VD));
        }
    }
}

extern "C" void kernel_launch(void* const* d_in, const int* in_sizes, int n_in,
                              void* d_out, int out_size, void* d_ws, size_t ws_size,
                              hipStream_t stream) {
    const float* enc  = (const float*)d_in[0];   // [B,T,D]
    const float* dec  = (const float*)d_in[1];   // [B,U,D]
    const float* W    = (const float*)d_in[2];   // [V,D]
    const float* bias = (const float*)d_in[3];   // [V]
    float* out  = (float*)d_out;                 // [B,T,U,V]

    float* encW = (float*)d_ws;                  // [B*T, V]  = 4 MB
    float* decW = encW + (size_t)BD * TD * VD;   // [B*U, V]  = 2 MB

    // enc GEMM: M = B*T = 1024 -> (1024/16)*(1024/64) = 1024 waves -> 128 blocks of 8 waves
    jointer_gemm_xwT<<<128, 256, 0, stream>>>(enc, W, bias, encW, BD * TD);
    // dec GEMM: M = B*U = 512  -> 512 waves -> 64 blocks
    jointer_gemm_xwT<<<64, 256, 0, stream>>>(dec, W, nullptr, decW, BD * UD);
    // broadcast add: B * (T/4) * (U/32) = 1024 blocks
    jointer_bcast_add<<<BD * (TD / 4) * (UD / 32), 256, 0, stream>>>(encW, decW, out);
}